// AnchorProcessor_8641474200313
// MI455X (gfx1250) — compile-verified
//
#include <hip/hip_runtime.h>
#include <math.h>

// YOLO-head decode for MI455X (gfx1250, wave32).
// Bandwidth-bound (~143 MB total @ 23.3 TB/s ~ 6 us). Uses the CDNA5 async
// global->LDS copy path (GLOBAL_LOAD_ASYNC_TO_LDS_B128 + ASYNCcnt) to
// double-buffer the 80 class channels per batch while reducing the previous
// batch's scores.

namespace {

constexpr int N_   = 8;            // batch
constexpr int A_   = 3;            // anchors
constexpr int CLS_ = 80;           // classes
constexpr int CH   = 5 + CLS_;     // 85 channels per anchor
constexpr int C_   = A_ * CH;      // 255
constexpr int Wd   = 128;
constexpr int Hd   = 128;
constexpr int HWc  = Wd * Hd;      // 16384
constexpr int PIX  = 128;          // pixels per block
constexpr int THREADS = 128;       // 4 wave32 waves
// bytes per stage = CLS_*PIX*4 = 40 KB; 16B chunks = 2560; per thread:
constexpr int CHUNKS = (CLS_ * PIX * 4 / 16) / THREADS;   // 20
constexpr int BUF = CLS_ * PIX;    // floats per LDS buffer (10240 = 40 KB)

// Issue the async copy of the 80 class channels of batch n into LDS buffer.
__device__ __forceinline__ void async_stage(const float* __restrict__ x,
                                            unsigned ldsbase, int buf,
                                            int n, int a, int p0, int tid) {
  // Uniform SGPR base: start of (n, anchor, ch=5) plane at this block's pixels.
  const float* gbase = x + (size_t)(n * C_ + a * CH + 5) * HWc + p0;
  const unsigned lbase = ldsbase + (unsigned)(buf * BUF * 4);
#pragma unroll
  for (int r = 0; r < CHUNKS; ++r) {
    const int chunk = r * THREADS + tid;        // 0..2559
    const int ch    = chunk >> 5;               // 32 x 16B chunks per channel
    const int off   = (chunk & 31) << 4;        // byte offset within 512B row
    const unsigned laddr = lbase + (unsigned)(ch * PIX * 4 + off);
    const unsigned goff  = (unsigned)(ch * HWc * 4 + off);
    // GVS form: LDS-dest VGPR, 32-bit per-lane offset VGPR, 64-bit SGPR base.
    asm volatile("global_load_async_to_lds_b128 %0, %1, %2"
                 :
                 : "v"(laddr), "v"(goff), "s"(gbase)
                 : "memory");
  }
}

} // namespace

__global__ __launch_bounds__(THREADS)
void yolo_head_kernel(const float* __restrict__ x, float* __restrict__ out) {
  extern __shared__ float sbuf[];     // 2 * BUF floats = 80 KB dynamic LDS
  const int tid = threadIdx.x;
  const int a   = blockIdx.y;         // anchor
  const int p0  = blockIdx.x * PIX;   // pixel base
  const int p   = p0 + tid;           // this thread's pixel

  // Low 32 bits of the flat LDS address == LDS byte offset (aperture layout).
  const unsigned ldsbase = (unsigned)(size_t)&sbuf[0];

  const float ax = (a == 0) ? 116.0f : (a == 1) ? 156.0f : 373.0f;
  const float ay = (a == 0) ? 90.0f  : (a == 1) ? 198.0f : 326.0f;
  const float gx = (float)(p & (Wd - 1));
  const float gy = (float)(p >> 7);

  // Prime the pipeline with batch 0.
  async_stage(x, ldsbase, 0, 0, a, p0, tid);

  float smax = -INFINITY;
  int   sarg = 0;

  for (int n = 0; n < N_; ++n) {
    const int b = n & 1;
    if (n + 1 < N_) {
      async_stage(x, ldsbase, b ^ 1, n + 1, a, p0, tid);
      // In-order completion: <=CHUNKS outstanding => all of stage n landed.
      asm volatile("s_wait_asynccnt %0" :: "i"(CHUNKS) : "memory");
    } else {
      asm volatile("s_wait_asynccnt 0" ::: "memory");
    }
    __syncthreads();   // all waves' slices of buffer b are now in LDS

    // Box channels 0..4: plain coalesced global loads (5/85 of traffic).
    const float* pb = x + (size_t)(n * C_ + a * CH) * HWc + p;
    const float t0  = pb[0];
    const float t1  = pb[1 * HWc];
    const float t2  = pb[2 * HWc];
    const float t3  = pb[3 * HWc];
    const float obj = pb[4 * HWc];

    float* ob = out + (size_t)(n * A_ * 6 + a * 6) * HWc + p;
    ob[0 * HWc] = 1.0f / (1.0f + __expf(-t0)) + gx;  // bx
    ob[1 * HWc] = 1.0f / (1.0f + __expf(-t1)) + gy;  // by
    ob[2 * HWc] = t2 * ax;                           // bw (raw * anchor)
    ob[3 * HWc] = t3 * ay;                           // bh

    // Score reduction over classes for this batch (flat index n*80 + c,
    // strict '>' keeps jnp.argmax first-occurrence semantics).
    const float* sb = &sbuf[b * BUF];
#pragma unroll
    for (int c = 0; c < CLS_; ++c) {
      const float s = sb[c * PIX + tid] * obj;   // conflict-free LDS read
      if (s > smax) { smax = s; sarg = n * CLS_ + c; }
    }
    __syncthreads();   // buffer b fully consumed before it is refilled
  }

  // Broadcast max/argmax to every batch element.
  const float farg = (float)sarg;
#pragma unroll
  for (int n = 0; n < N_; ++n) {
    float* ob = out + (size_t)(n * A_ * 6 + a * 6 + 4) * HWc + p;
    ob[0]   = smax;
    ob[HWc] = farg;
  }
}

extern "C" void kernel_launch(void* const* d_in, const int* in_sizes, int n_in,
                              void* d_out, int out_size, void* d_ws, size_t ws_size,
                              hipStream_t stream) {
  (void)in_sizes; (void)n_in; (void)out_size; (void)d_ws; (void)ws_size;
  const float* x = (const float*)d_in[0];
  float* out = (float*)d_out;

  dim3 grid(HWc / PIX, A_);          // 128 x 3 = 384 blocks
  dim3 block(THREADS);               // 128 threads = 4 wave32 waves
  const size_t shmem = 2u * BUF * sizeof(float);   // 80 KB double buffer
  hipLaunchKernelGGL(yolo_head_kernel, grid, block, shmem, stream, x, out);
}